// CP_ABR_Plus_18287970746774
// MI455X (gfx1250) — compile-verified
//
#include <hip/hip_runtime.h>
#include <hip/hip_bf16.h>
#include <stdint.h>

typedef __attribute__((ext_vector_type(2))) float v2f;
typedef __attribute__((ext_vector_type(8))) float v8f;

// ---------------------------------------------------------------------------
// Degree / normalization kernels
// ---------------------------------------------------------------------------
__global__ void deg_init_kernel(float* __restrict__ deg, int n) {
    int i = blockIdx.x * blockDim.x + threadIdx.x;
    if (i < n) deg[i] = 1.0f;               // self-loop contributes 1 to every node
}

__global__ void deg_count_kernel(const long long* __restrict__ dst,
                                 float* __restrict__ deg, long long e) {
    long long i = (long long)blockIdx.x * blockDim.x + threadIdx.x;
    if (i < e) atomicAdd(&deg[(int)dst[i]], 1.0f);
}

__global__ void deg_rsqrt_kernel(float* __restrict__ deg, int n) {
    int i = blockIdx.x * blockDim.x + threadIdx.x;
    if (i < n) deg[i] = rsqrtf(deg[i]);     // deg >= 1 always (self-loop)
}

// ---------------------------------------------------------------------------
// fp32 WMMA GEMM: D[M x Ncols] = A[M x K] @ B[K x Ncols]
// One wave computes one 16x16 tile with V_WMMA_F32_16X16X4_F32 (K-step 4).
// Requires M % 16 == 0, K % 4 == 0, Ncols % 16 == 0 (true here).
// ---------------------------------------------------------------------------
__global__ void __launch_bounds__(256)
gemm_f32_wmma_kernel(const float* __restrict__ A, const float* __restrict__ B,
                     float* __restrict__ D, int M, int K, int Ncols) {
    const int lane     = threadIdx.x & 31;
    const int wave     = blockIdx.x * (blockDim.x >> 5) + (threadIdx.x >> 5);
    const int colTiles = Ncols >> 4;
    const int rowTile  = wave / colTiles;
    const int colTile  = wave - rowTile * colTiles;
    if (rowTile * 16 >= M) return;          // uniform per wave

    const int mr = lane & 15;               // A row / B,N col within tile
    const int kg = (lane >> 4) << 1;        // k sub-group: 0 or 2 (ISA A/B layout)

    const float* __restrict__ arow = A + (size_t)(rowTile * 16 + mr) * K + kg;
    const float* __restrict__ bptr = B + (size_t)kg * Ncols + (colTile * 16 + mr);

    v8f acc = {};
    for (int k0 = 0; k0 < K; k0 += 4) {
        v2f a, b;
        a.x = arow[k0];                     // A[m][k0+kg]   (contiguous pair)
        a.y = arow[k0 + 1];                 // A[m][k0+kg+1]
        b.x = bptr[(size_t)k0 * Ncols];     // B[k0+kg][n]
        b.y = bptr[(size_t)(k0 + 1) * Ncols];
        acc = __builtin_amdgcn_wmma_f32_16x16x4_f32(
            /*neg_a=*/false, a, /*neg_b=*/false, b,
            /*c_mod=*/(short)0, acc, /*reuse_a=*/false, /*reuse_b=*/false);
    }

    // D layout: VGPR v <-> row (v + 8*(lane>=16)), col = lane&15
    const int row0 = rowTile * 16 + ((lane >> 4) << 3);
    const int col  = colTile * 16 + mr;
    float* __restrict__ dptr = D + (size_t)row0 * Ncols + col;
#pragma unroll
    for (int v = 0; v < 8; v++) dptr[(size_t)v * Ncols] = acc[v];
}

// ---------------------------------------------------------------------------
// Edge aggregation: agg[d] += xw[s] * (dinv[s]*dinv[d]).
// One wave per edge; each lane owns a float4 slice of the 128 features.
// ---------------------------------------------------------------------------
__global__ void __launch_bounds__(256)
aggregate_kernel(const float* __restrict__ xw, const float* __restrict__ dinv,
                 const long long* __restrict__ src, const long long* __restrict__ dst,
                 float* __restrict__ agg, long long e, int hid) {
    long long tid = (long long)blockIdx.x * blockDim.x + threadIdx.x;
    long long ed  = tid >> 5;               // edge index (uniform per wave)
    int lane      = (int)(tid & 31);
    if (ed >= e) return;
    int s = (int)src[ed];
    int d = (int)dst[ed];
    float w = dinv[s] * dinv[d];
    const float4* __restrict__ xs = (const float4*)(xw + (size_t)s * hid);
    float4 v = xs[lane];
    float* __restrict__ ap = agg + (size_t)d * hid + lane * 4;
    atomicAdd(ap + 0, v.x * w);
    atomicAdd(ap + 1, v.y * w);
    atomicAdd(ap + 2, v.z * w);
    atomicAdd(ap + 3, v.w * w);
}

// ---------------------------------------------------------------------------
// h = relu(agg + xw*dinv^2 + b)    (adds the self-loop message + bias)
// ---------------------------------------------------------------------------
__global__ void finalize_relu_kernel(const float* __restrict__ agg,
                                     const float* __restrict__ xw,
                                     const float* __restrict__ dinv,
                                     const float* __restrict__ b,
                                     float* __restrict__ h, int n, int hid) {
    long long tid = (long long)blockIdx.x * blockDim.x + threadIdx.x;
    if (tid >= (long long)n * hid) return;
    int i = (int)(tid / hid);
    int f = (int)(tid - (long long)i * hid);
    float di = dinv[i];
    float v  = agg[tid] + xw[tid] * di * di + b[f];
    h[tid]   = fmaxf(v, 0.0f);
}

// ---------------------------------------------------------------------------
// Final fused kernel: one wave per node.
//   h2 = relu(agg2 + xw2*dinv^2 + b2)
//   ha = alpha*h2 + (1-alpha)*h1
//   out = 0.5*(h1@Wc + bc) + 0.5*(ha@Wf + bf)
// ---------------------------------------------------------------------------
__global__ void __launch_bounds__(256)
final_out_kernel(const float* __restrict__ h1, const float* __restrict__ xw2,
                 const float* __restrict__ agg2, const float* __restrict__ dinv,
                 const float* __restrict__ hnode, const float* __restrict__ b2,
                 const float* __restrict__ Wc, const float* __restrict__ bc,
                 const float* __restrict__ Wf, const float* __restrict__ bf,
                 float* __restrict__ out, int n, int hid) {
    int node = blockIdx.x * (blockDim.x >> 5) + (threadIdx.x >> 5);
    int lane = threadIdx.x & 31;
    if (node >= n) return;                  // uniform per wave

    float di    = dinv[node];
    float selfw = di * di;
    float alpha = hnode[node];
    const float* __restrict__ h1p = h1   + (size_t)node * hid;
    const float* __restrict__ x2p = xw2  + (size_t)node * hid;
    const float* __restrict__ a2p = agg2 + (size_t)node * hid;

    float c0 = 0.f, c1 = 0.f, f0 = 0.f, f1 = 0.f;
#pragma unroll
    for (int k = 0; k < 4; k++) {
        int f = lane * 4 + k;
        float h1v = h1p[f];
        float h2v = fmaxf(a2p[f] + x2p[f] * selfw + b2[f], 0.0f);
        float hav = alpha * h2v + (1.0f - alpha) * h1v;
        c0 += h1v * Wc[f * 2 + 0];
        c1 += h1v * Wc[f * 2 + 1];
        f0 += hav * Wf[f * 2 + 0];
        f1 += hav * Wf[f * 2 + 1];
    }
#pragma unroll
    for (int off = 16; off >= 1; off >>= 1) {
        c0 += __shfl_down(c0, off, 32);
        c1 += __shfl_down(c1, off, 32);
        f0 += __shfl_down(f0, off, 32);
        f1 += __shfl_down(f1, off, 32);
    }
    if (lane == 0) {
        out[(size_t)node * 2 + 0] = 0.5f * (c0 + bc[0]) + 0.5f * (f0 + bf[0]);
        out[(size_t)node * 2 + 1] = 0.5f * (c1 + bc[1]) + 0.5f * (f1 + bf[1]);
    }
}

// ---------------------------------------------------------------------------
extern "C" void kernel_launch(void* const* d_in, const int* in_sizes, int n_in,
                              void* d_out, int out_size, void* d_ws, size_t ws_size,
                              hipStream_t stream) {
    const float*     x   = (const float*)d_in[0];
    const long long* ei  = (const long long*)d_in[1];   // int64 edge_index (2,E)
    const float*     hn  = (const float*)d_in[2];
    const float*     W1  = (const float*)d_in[3];
    const float*     b1  = (const float*)d_in[4];
    const float*     W2  = (const float*)d_in[5];
    const float*     b2  = (const float*)d_in[6];
    const float*     Wc  = (const float*)d_in[7];
    const float*     bc  = (const float*)d_in[8];
    const float*     Wf  = (const float*)d_in[9];
    const float*     bf  = (const float*)d_in[10];
    float*           out = (float*)d_out;

    const int       Nn  = in_sizes[2];          // 100000
    const long long Ee  = (long long)in_sizes[1] / 2;   // 1.6M
    const int       HID = in_sizes[4];          // 128
    const int       IND = in_sizes[0] / Nn;     // 256

    const long long* src = ei;
    const long long* dst = ei + Ee;

    float* ws   = (float*)d_ws;
    float* dinv = ws;                           // N      (deg -> rsqrt in place)
    float* bufA = dinv + Nn;                    // N*HID  (xw1, later xw2)
    float* bufB = bufA + (size_t)Nn * HID;      // N*HID  (h1)
    float* bufC = bufB + (size_t)Nn * HID;      // N*HID  (agg scratch)

    const size_t featBytes = (size_t)Nn * HID * sizeof(float);

    // 1) symmetric normalization coefficients
    deg_init_kernel<<<(Nn + 255) / 256, 256, 0, stream>>>(dinv, Nn);
    deg_count_kernel<<<(int)((Ee + 255) / 256), 256, 0, stream>>>(dst, dinv, Ee);
    deg_rsqrt_kernel<<<(Nn + 255) / 256, 256, 0, stream>>>(dinv, Nn);

    const int tiles = (Nn / 16) * (HID / 16);   // 16x16 output tiles, 8 waves/block
    const int gemmBlocks = (tiles + 7) / 8;
    const int aggBlocks  = (int)((Ee * 32 + 255) / 256);
    const int eltBlocks  = (int)(((long long)Nn * HID + 255) / 256);

    // 2) layer 1: xw1 = x @ W1 ; aggregate ; h1 = relu(... + b1)
    gemm_f32_wmma_kernel<<<gemmBlocks, 256, 0, stream>>>(x, W1, bufA, Nn, IND, HID);
    hipMemsetAsync(bufC, 0, featBytes, stream);
    aggregate_kernel<<<aggBlocks, 256, 0, stream>>>(bufA, dinv, src, dst, bufC, Ee, HID);
    finalize_relu_kernel<<<eltBlocks, 256, 0, stream>>>(bufC, bufA, dinv, b1, bufB, Nn, HID);

    // 3) layer 2: xw2 = h1 @ W2 ; aggregate ; fused h2/blend/heads -> out
    gemm_f32_wmma_kernel<<<gemmBlocks, 256, 0, stream>>>(bufB, W2, bufA, Nn, HID, HID);
    hipMemsetAsync(bufC, 0, featBytes, stream);
    aggregate_kernel<<<aggBlocks, 256, 0, stream>>>(bufA, dinv, src, dst, bufC, Ee, HID);
    final_out_kernel<<<(Nn + 7) / 8, 256, 0, stream>>>(bufB, bufA, bufC, dinv, hn,
                                                       b2, Wc, bc, Wf, bf, out, Nn, HID);
}